// GARCHLayer_63393717289152
// MI455X (gfx1250) — compile-verified
//
#include <hip/hip_runtime.h>
#include <hip/hip_bf16.h>

// Problem constants (match reference)
#define BATCH 16
#define SEQ   4096
#define DIM   512
#define DH    256
#define SEGS  64     // time segments per sequence
#define SEGLEN 64    // steps per segment
#define MTILE 64     // rows per block in fused GEMM

typedef float v2f __attribute__((ext_vector_type(2)));
typedef float v8f __attribute__((ext_vector_type(8)));

// ---------------------------------------------------------------------------
// Pass A: segmented GARCH partial scans.
// One thread per (b, seg, d). Writes partial cond_var into out[b,t,d] and the
// segment-end partial into segE[(b*SEGS+s)*DIM + d].
// ---------------------------------------------------------------------------
__global__ __launch_bounds__(256) void garch_scan_seg(
    const float* __restrict__ ret,
    const float* __restrict__ alpha_p,
    const float* __restrict__ beta_p,
    const float* __restrict__ omega_p,
    float* __restrict__ cv_out,     // d_out used as staging for partial cv
    float* __restrict__ segE)
{
    int tid = blockIdx.x * blockDim.x + threadIdx.x;   // 0 .. 16*64*512-1
    int d = tid & (DIM - 1);
    int s = (tid >> 9) & (SEGS - 1);
    int b = tid >> 15;

    float alpha = alpha_p[0];
    float beta  = beta_p[0];
    float omega = omega_p[0];

    const float* rbase = ret    + (size_t)b * SEQ * DIM + d;
    float*       obase = cv_out + (size_t)b * SEQ * DIM + d;

    int t0 = s * SEGLEN;
    float cv;
    int kstart;
    if (s == 0) {
        cv = 0.01f;                 // INIT_VAR, exact
        obase[0] = cv;
        kstart = 1;
    } else {
        cv = 0.0f;                  // carry-in applied later (linear recurrence)
        kstart = 0;
    }
    for (int k = kstart; k < SEGLEN; ++k) {
        int t = t0 + k;
        float r = rbase[(size_t)(t - 1) * DIM];
        cv = omega + alpha * r * r + beta * cv;
        obase[(size_t)t * DIM] = cv;
    }
    segE[((size_t)(b * SEGS + s)) * DIM + d] = cv;
}

// ---------------------------------------------------------------------------
// Pass B: combine segment carries (in place).
// E[s] = end_partial[s] + beta^64 * E[s-1];  E[0] is already exact.
// One thread per (b, d); loads/stores coalesced across d for each s.
// ---------------------------------------------------------------------------
__global__ __launch_bounds__(256) void garch_combine(
    const float* __restrict__ beta_p,
    float* __restrict__ segE)
{
    int tid = blockIdx.x * blockDim.x + threadIdx.x;   // 0 .. 8191
    int d = tid & (DIM - 1);
    int b = tid >> 9;

    float beta = beta_p[0];
    float b64v = beta;
    for (int i = 0; i < 63; ++i) b64v *= beta;         // beta^64

    float E = 0.0f;
    for (int s = 0; s < SEGS; ++s) {
        size_t idx = ((size_t)(b * SEGS + s)) * DIM + d;
        float pe = segE[idx];
        E = (s == 0) ? pe : pe + b64v * E;
        segE[idx] = E;                                  // now true cv at t = 64*s+63
    }
}

// ---------------------------------------------------------------------------
// Pass C: fused  relu(x@W1+b1) @ W2 + b2 -> softplus -> * cond_var  per 64-row
// tile, using V_WMMA_F32_16X16X4_F32. 8 waves per block; each wave owns a
// 16x128 strip of the output tile (8 x v8f accumulators).
// ---------------------------------------------------------------------------
__global__ __launch_bounds__(256) void garch_mlp_fused(
    const float* __restrict__ x,
    const float* __restrict__ W1,
    const float* __restrict__ b1,
    const float* __restrict__ W2,
    const float* __restrict__ b2,
    const float* __restrict__ segE,
    const float* __restrict__ beta_p,
    float* __restrict__ out)        // holds partial cv on entry; final result on exit
{
    __shared__ float h_s[MTILE * DH];   // 64 KB: relu(x@W1+b1) tile

    const int m0   = blockIdx.x * MTILE;       // first row of tile (row = b*SEQ + t)
    const int b    = m0 / SEQ;
    const int t0   = m0 % SEQ;                 // multiple of 64 -> whole tile in one segment
    const int seg  = t0 / SEGLEN;

    const int lane  = threadIdx.x & 31;
    const int wave  = threadIdx.x >> 5;
    const int r     = wave & 3;                // row-tile (16 rows each)
    const int cHalf = wave >> 2;               // column half
    const int half  = lane >> 4;               // K-pair select within fragment
    const int ml    = lane & 15;               // M (A) or N (B/C) index

    const float betav = beta_p[0];
    const v8f vzero = {0.f, 0.f, 0.f, 0.f, 0.f, 0.f, 0.f, 0.f};

    // ---------------- Phase 1: h = relu(x @ W1 + b1) -> LDS ----------------
    v8f acc[8];
#pragma unroll
    for (int j = 0; j < 8; ++j) acc[j] = vzero;

    const float* xrow = x + (size_t)(m0 + 16 * r + ml) * DIM;
    for (int k0 = 0; k0 < DIM; k0 += 4) {
        const int kk = k0 + 2 * half;
        v2f a = *(const v2f*)(xrow + kk);      // A frag: lane M=ml, K = kk, kk+1
#pragma unroll
        for (int j = 0; j < 8; ++j) {
            const int col = cHalf * 128 + j * 16 + ml;
            v2f bf;                            // B frag: lane N=ml (via col), K = kk, kk+1
            bf.x = W1[(size_t)kk * DH + col];
            bf.y = W1[(size_t)(kk + 1) * DH + col];
            acc[j] = __builtin_amdgcn_wmma_f32_16x16x4_f32(
                false, a, false, bf, (short)0, acc[j], false, false);
        }
    }
#pragma unroll
    for (int j = 0; j < 8; ++j) {
        const int col = cHalf * 128 + j * 16 + ml;
        const float bias = b1[col];
#pragma unroll
        for (int v = 0; v < 8; ++v) {
            const int row = 16 * r + 8 * half + v;   // C layout: M = v + 8*half
            float hv = acc[j][v] + bias;
            h_s[row * DH + col] = hv > 0.0f ? hv : 0.0f;
        }
    }
    __syncthreads();

    // ------- Phase 2: out = softplus(h @ W2 + b2) * cond_var (2 col passes) -------
    // beta^{row_local+1} base for v=0 of this wave's rows
    float bp0 = 1.0f;
    {
        const int e0 = 16 * r + 8 * half + 1;
        for (int i = 0; i < e0; ++i) bp0 *= betav;
    }

    for (int cb = 0; cb < 2; ++cb) {
        v8f acc2[8];
#pragma unroll
        for (int j = 0; j < 8; ++j) acc2[j] = vzero;

        const float* hrow = &h_s[(16 * r + ml) * DH];
        for (int k0 = 0; k0 < DH; k0 += 4) {
            const int kk = k0 + 2 * half;
            v2f a = *(const v2f*)(hrow + kk);          // ds_load_b64 from LDS
#pragma unroll
            for (int j = 0; j < 8; ++j) {
                const int col = cb * 256 + cHalf * 128 + j * 16 + ml;
                v2f bf;
                bf.x = W2[(size_t)kk * DIM + col];
                bf.y = W2[(size_t)(kk + 1) * DIM + col];
                acc2[j] = __builtin_amdgcn_wmma_f32_16x16x4_f32(
                    false, a, false, bf, (short)0, acc2[j], false, false);
            }
        }

#pragma unroll
        for (int j = 0; j < 8; ++j) {
            const int col = cb * 256 + cHalf * 128 + j * 16 + ml;
            const float bias = b2[col];
            const float Ecarry = (seg > 0)
                ? segE[((size_t)(b * SEGS + (seg - 1))) * DIM + col]
                : 0.0f;
            float bp = bp0;                            // beta^{row_local+1}
#pragma unroll
            for (int v = 0; v < 8; ++v) {
                const int row_local = 16 * r + 8 * half + v;
                const size_t om = (size_t)(m0 + row_local) * DIM + col;
                const float pre = acc2[j][v] + bias;
                // jax.nn.softplus: max(x,0) + log1p(exp(-|x|))
                const float sp = fmaxf(pre, 0.0f) + log1pf(expf(-fabsf(pre)));
                const float cv = out[om] + bp * Ecarry;  // partial + beta^{k+1} * carry
                out[om] = sp * cv;
                bp *= betav;
            }
        }
    }
}

// ---------------------------------------------------------------------------
// Host-side launch. Inputs: x, returns, alpha, beta, omega, W1, b1, W2, b2.
// ws layout: segE = float[BATCH*SEGS*DIM]  (2 MB)
// ---------------------------------------------------------------------------
extern "C" void kernel_launch(void* const* d_in, const int* in_sizes, int n_in,
                              void* d_out, int out_size, void* d_ws, size_t ws_size,
                              hipStream_t stream) {
    const float* x     = (const float*)d_in[0];
    const float* ret   = (const float*)d_in[1];
    const float* alpha = (const float*)d_in[2];
    const float* beta  = (const float*)d_in[3];
    const float* omega = (const float*)d_in[4];
    const float* W1    = (const float*)d_in[5];
    const float* b1    = (const float*)d_in[6];
    const float* W2    = (const float*)d_in[7];
    const float* b2    = (const float*)d_in[8];
    float* out  = (float*)d_out;
    float* segE = (float*)d_ws;

    // Pass A: 16*64*512 threads = 2048 blocks x 256
    garch_scan_seg<<<(BATCH * SEGS * DIM) / 256, 256, 0, stream>>>(
        ret, alpha, beta, omega, out, segE);

    // Pass B: 16*512 threads = 32 blocks x 256
    garch_combine<<<(BATCH * DIM) / 256, 256, 0, stream>>>(beta, segE);

    // Pass C: one block per 64 output rows -> 65536/64 = 1024 blocks
    garch_mlp_fused<<<(BATCH * SEQ) / MTILE, 256, 0, stream>>>(
        x, W1, b1, W2, b2, segE, beta, out);
}